// MixedScore_MultiHeadAttention_5935644803612
// MI455X (gfx1250) — compile-verified
//
#include <hip/hip_runtime.h>

// MixedScore multi-head attention for MI455X (gfx1250, wave32).
// B=4, H=8, N=512, D=16. One wave (32 threads) handles one 16-row query
// tile of one (b,h): S = QK^T/4 via V_WMMA_F32_16X16X4_F32, per-element
// 2->16->1 ReLU MLP in VALU (wave-uniform SGPR coefficients -> v_fma_f32
// with scalar operands + VOPD dual-issue), two-pass softmax staged through
// LDS, then O = P*V via WMMA again.
//
// NOTE: a V_PK_FMA_F32 variant was tried and rejected: broadcasting the 64
// wave-uniform coefficients into float2 operands pushed them from SGPRs into
// 64 VGPR pairs, exceeding the 256-VGPR window and flooding the stream with
// s_set_vgpr_msb (salu 146 -> 723, total 738 -> 1721). Scalar-operand FMAs
// keep coefficients in SGPRs and dual-issue instead.

#define HN    8
#define QD    16
#define SEQ   512
#define BATCH 4
#define JT    (SEQ / 16)   // 32 column tiles
#define LDAP  516          // padded LDS row pitch (words); 8*516 % 64 = 32 -> no bank clash between lane halves

typedef float v2f __attribute__((ext_vector_type(2)));
typedef float v8f __attribute__((ext_vector_type(8)));

// Only the device pass declares AMDGCN builtins; the host pass must not see this check.
#if defined(__HIP_DEVICE_COMPILE__) && !__has_builtin(__builtin_amdgcn_wmma_f32_16x16x4_f32)
#error "gfx1250 f32 WMMA builtin not available on device pass"
#endif

__global__ __launch_bounds__(32) void msmha_kernel(
    const float* __restrict__ q, const float* __restrict__ k,
    const float* __restrict__ v, const float* __restrict__ cost,
    const float* __restrict__ w1, const float* __restrict__ b1,
    const float* __restrict__ w2, const float* __restrict__ b2,
    float* __restrict__ out)
{
    __shared__ float sm[16 * LDAP];  // mixed scores: 16 query rows x 512 cols (padded)
    __shared__ float stats[32];      // [0..15] row max, [16..31] row 1/sum

    const int lane = threadIdx.x;    // wave32
    const int g = lane >> 4;         // lane half (0/1)
    const int c = lane & 15;         // position within half

    const int blk = blockIdx.x;      // 1024 = B * H * JT
    const int it  = blk & (JT - 1);
    const int h   = (blk >> 5) & (HN - 1);
    const int b   = blk >> 8;
    const int i0  = it * 16;

    const float* qp = q + (size_t)((b * HN + h) * SEQ) * QD;
    const float* kp = k + (size_t)((b * HN + h) * SEQ) * QD;
    const float* vp = v + (size_t)((b * HN + h) * SEQ) * QD;
    const float* cp = cost + (size_t)b * SEQ * SEQ;

    // Per-head MLP coefficients: wave-uniform -> scalar loads / SGPRs.
    float am[16], cm[16], bmv[16], wmv[16];
#pragma unroll
    for (int m = 0; m < 16; ++m) {
        am[m]  = w1[h * 32 + m];       // mix1_weight[h,0,m]
        cm[m]  = w1[h * 32 + 16 + m];  // mix1_weight[h,1,m]
        bmv[m] = b1[h * 16 + m];
        wmv[m] = w2[h * 16 + m];
    }
    const float bias2 = b2[h];

    // Q tile in WMMA A-layout (16x4 f32 chunks): lane holds row i0+c,
    // VGPR v of chunk cc holds K = 4*cc + v + 2*g.  Fold the 1/sqrt(D)=0.25.
    v2f qa[4];
#pragma unroll
    for (int cc = 0; cc < 4; ++cc) {
        const float* p = qp + (i0 + c) * QD + cc * 4 + 2 * g;
        qa[cc][0] = p[0] * 0.25f;
        qa[cc][1] = p[1] * 0.25f;
    }

    // ---------------- Pass A: mixed scores -> LDS ----------------
#pragma unroll 1
    for (int jt = 0; jt < JT; ++jt) {
        const int j0 = jt * 16;

        // K tile in B-layout: lane -> column j0+c, VGPR v of chunk cc -> K = 4*cc + v + 2*g.
        v2f kb[4];
#pragma unroll
        for (int cc = 0; cc < 4; ++cc) {
            const float* p = kp + (j0 + c) * QD + cc * 4 + 2 * g;
            kb[cc][0] = p[0];
            kb[cc][1] = p[1];
        }

        v8f s = {};
#pragma unroll
        for (int cc = 0; cc < 4; ++cc)
            s = __builtin_amdgcn_wmma_f32_16x16x4_f32(false, qa[cc], false, kb[cc],
                                                      (short)0, s, false, false);

        // cost in the same C/D layout: VGPR r -> row i0 + r + 8*g, col j0 + c.
        float cr[8];
#pragma unroll
        for (int r = 0; r < 8; ++r)
            cr[r] = cp[(size_t)(i0 + r + 8 * g) * SEQ + j0 + c];

        // Per-element MLP: mixed = b2 + sum_m w2[m]*relu(a_m*dot + c_m*cost + b_m)
        float acc[8];
#pragma unroll
        for (int r = 0; r < 8; ++r) acc[r] = bias2;
#pragma unroll
        for (int m = 0; m < 16; ++m) {
            const float a_ = am[m], c_ = cm[m], b_ = bmv[m], w_ = wmv[m];
#pragma unroll
            for (int r = 0; r < 8; ++r) {
                float t = fmaf(a_, s[r], b_);
                t = fmaf(c_, cr[r], t);
                t = fmaxf(t, 0.0f);
                acc[r] = fmaf(w_, t, acc[r]);
            }
        }

#pragma unroll
        for (int r = 0; r < 8; ++r)
            sm[(r + 8 * g) * LDAP + j0 + c] = acc[r];
    }
    __syncthreads();

    // ---------------- Pass B: row softmax stats ----------------
    {
        const int row = c;                              // lane half g scans half the row
        const float* rp = &sm[row * LDAP + g * 256];
        float mx = -3.402823466e38f;
#pragma unroll 1
        for (int jj = 0; jj < 64; ++jj) {
            float4 x = ((const float4*)rp)[jj];
            mx = fmaxf(mx, fmaxf(fmaxf(x.x, x.y), fmaxf(x.z, x.w)));
        }
        float sum = 0.0f;
#pragma unroll 1
        for (int jj = 0; jj < 64; ++jj) {
            float4 x = ((const float4*)rp)[jj];
            sum += __expf(x.x - mx) + __expf(x.y - mx) +
                   __expf(x.z - mx) + __expf(x.w - mx);
        }
        const float mo = __shfl_xor(mx, 16, 32);
        const float so = __shfl_xor(sum, 16, 32);
        const float M  = fmaxf(mx, mo);
        const float S  = sum * __expf(mx - M) + so * __expf(mo - M);
        if (g == 0) {
            stats[row]      = M;
            stats[16 + row] = 1.0f / S;
        }
    }
    __syncthreads();

    // ---------------- Pass C: O = softmax(mixed) * V via WMMA ----------------
    const float m_a = stats[c];   // A-layout: lane -> query row c
    v8f o = {};
#pragma unroll 1
    for (int jt = 0; jt < JT; ++jt) {
        const int j0 = jt * 16;
        v2f pa[4], vb[4];
#pragma unroll
        for (int cc = 0; cc < 4; ++cc) {
            // P chunk in A-layout straight out of row-major LDS (free transpose):
            // lane -> row c, K = j0 + 4*cc + 2*g + {0,1}
            const v2f x = *(const v2f*)&sm[c * LDAP + j0 + cc * 4 + 2 * g];
            pa[cc][0] = __expf(x[0] - m_a);
            pa[cc][1] = __expf(x[1] - m_a);
            // V chunk in B-layout: lane -> d = c, K(row of V) = j0 + 4*cc + v + 2*g
            vb[cc][0] = vp[(size_t)(j0 + cc * 4 + 0 + 2 * g) * QD + c];
            vb[cc][1] = vp[(size_t)(j0 + cc * 4 + 1 + 2 * g) * QD + c];
        }
#pragma unroll
        for (int cc = 0; cc < 4; ++cc)
            o = __builtin_amdgcn_wmma_f32_16x16x4_f32(false, pa[cc], false, vb[cc],
                                                      (short)0, o, false, false);
    }

    // O is in C/D layout: VGPR r -> query row i0 + r + 8*g, lane -> d = c.
    // Output layout: out[b, n, h*16 + d].
#pragma unroll
    for (int r = 0; r < 8; ++r) {
        const int row = r + 8 * g;
        const float invl = stats[16 + row];
        out[(size_t)(b * SEQ + i0 + row) * (HN * QD) + h * QD + c] = o[r] * invl;
    }
}

extern "C" void kernel_launch(void* const* d_in, const int* in_sizes, int n_in,
                              void* d_out, int out_size, void* d_ws, size_t ws_size,
                              hipStream_t stream)
{
    const float* q    = (const float*)d_in[0];
    const float* k    = (const float*)d_in[1];
    const float* v    = (const float*)d_in[2];
    const float* cost = (const float*)d_in[3];
    const float* w1   = (const float*)d_in[4];
    const float* b1   = (const float*)d_in[5];
    const float* w2   = (const float*)d_in[6];
    const float* b2   = (const float*)d_in[7];
    float* out = (float*)d_out;

    msmha_kernel<<<dim3(BATCH * HN * JT), dim3(32), 0, stream>>>(
        q, k, v, cost, w1, b1, w2, b2, out);
}